// AutoCRF_50337016709512
// MI455X (gfx1250) — compile-verified
//
#include <hip/hip_runtime.h>
#include <hip/hip_bf16.h>
#include <cstdint>

// Problem constants (from reference): B=1024, T=512, L=48
#define BSZ 1024
#define TSZ 512
#define LSZ 48
#define LPAD 49              // padded LDS stride (48 % 64 banks -> conflicts; 49 is coprime)
#define VROWS 8              // batch rows per viterbi block
#define BT_ROWS 4            // batch rows per backtrace block
#define BT_CH 32             // backtrace chunk (time steps staged in LDS)

typedef float v2f __attribute__((ext_vector_type(2)));
typedef float v8f __attribute__((ext_vector_type(8)));
typedef unsigned int v4u __attribute__((ext_vector_type(4)));
typedef int v4i __attribute__((ext_vector_type(4)));
typedef int v8i __attribute__((ext_vector_type(8)));

#if defined(__has_builtin)
#if __has_builtin(__builtin_amdgcn_tensor_load_to_lds) && \
    __has_builtin(__builtin_amdgcn_s_wait_tensorcnt)
#define USE_TDM 1
#endif
#endif
#ifndef USE_TDM
#define USE_TDM 0
#endif

#if USE_TDM
// ---------------------------------------------------------------------------
// Issue one TDM 2D tile load: 16 rows x 48 f32, row stride T*L elements,
// global tile base = gaddr, destination LDS byte offset = lds_off.
// D# layout per CDNA5 ISA ch.8 (group0: flags/lds/global+type=2;
// group1: data_size=4B, dims/strides; groups 2-3 zero for a 2D tensor).
// ---------------------------------------------------------------------------
__device__ __forceinline__ void tdm_load_tile(uint32_t lds_off, const float* gaddr) {
  uint64_t ga = (uint64_t)(uintptr_t)gaddr;
  v4u g0;
  g0[0] = 1u;                                        // count=1, is_restore=0
  g0[1] = lds_off;                                   // lds_addr (bytes)
  g0[2] = (uint32_t)(ga & 0xffffffffu);              // global_addr[31:0]
  g0[3] = (uint32_t)((ga >> 32) & 0x01ffffffu)       // global_addr[56:32]
          | (2u << 30);                              // type = 2 ("image")
  v8i g1;
  g1[0] = (int)(2u << 16);        // workgroup_mask=0 | data_size=2 (4 bytes)
  g1[1] = (int)(48u << 16);       // atomic_barrier_addr=0 | tensor_dim0.lo16 = 48
  g1[2] = (int)(16u << 16);       // tensor_dim0.hi16=0 | tensor_dim1.lo16 = 16
  g1[3] = (int)(48u << 16);       // tensor_dim1.hi16=0 | tile_dim0 = 48
  g1[4] = 16;                     // tile_dim1 = 16 | tile_dim2 = 0
  g1[5] = (int)(TSZ * LSZ);       // tensor_dim0_stride.lo32 = 24576 elements
  g1[6] = 0;                      // stride.hi16 | tensor_dim1_stride.lo16
  g1[7] = 0;
  v4i z4 = {0, 0, 0, 0};
#if defined(__clang_major__) && (__clang_major__ >= 23)
  v8i z8 = {0, 0, 0, 0, 0, 0, 0, 0};
  __builtin_amdgcn_tensor_load_to_lds(g0, g1, z4, z4, z8, 0);
#else
  __builtin_amdgcn_tensor_load_to_lds(g0, g1, z4, z4, 0);
#endif
}
#endif // USE_TDM

// ---------------------------------------------------------------------------
// Kernel A: CRF forward recurrence via exp-space f32 WMMA GEMM.
// One wave (32 threads) owns 16 batch rows. Per step:
//   m[b]      = max_i alpha[b][i]
//   P[b][i]   = exp(alpha[b][i] - m[b])
//   Z         = P @ expT                      (WMMA f32 16x16x4, K=48)
//   alpha'[b][j] = feats[b][t][j] + m[b] + log(Z[b][j])   (masked)
// expT B-fragments hoisted into registers for the entire T loop.
// feats tiles staged by the Tensor Data Mover, double-buffered one step ahead
// (s_wait_tensorcnt 1 overlaps DMA of t+1 with WMMA of t).
// ---------------------------------------------------------------------------
__global__ __launch_bounds__(32) void crf_forward_wmma(
    const float* __restrict__ feats, const unsigned char* __restrict__ mask,
    const float* __restrict__ trans, const float* __restrict__ start_t,
    const float* __restrict__ end_t, float* __restrict__ logZ)
{
  __shared__ float expT[LSZ][LPAD];
  __shared__ float alpha[16][LPAD];
  __shared__ float P[16][LPAD];
  __shared__ __align__(16) float tileT[2][16][LSZ];  // TDM-written (unpadded rows)
  __shared__ float mrow[16];
  __shared__ unsigned char mk[16];

  const int lane = threadIdx.x;
  const int b0 = blockIdx.x * 16;
  const float* gbase = feats + (size_t)b0 * TSZ * LSZ;

  // exp(T) once per block
  for (int idx = lane; idx < LSZ * LSZ; idx += 32) {
    int i = idx / LSZ, j = idx % LSZ;
    expT[i][j] = __expf(trans[idx]);
  }
  __syncthreads();

  // Hoist B-matrix fragments into registers (time-invariant).
  //   lanes 0-15: N=lane,    v.x=K0, v.y=K0+1 with K0 = 4*kt
  //   lanes 16-31: N=lane-16, K0 = 4*kt + 2
  const int nfrag = lane & 15;
  const int khalf = (lane >> 4) * 2;
  v2f Bf[12][3];
#pragma unroll
  for (int kt = 0; kt < 12; ++kt)
#pragma unroll
    for (int nt = 0; nt < 3; ++nt) {
      Bf[kt][nt].x = expT[kt * 4 + khalf][nt * 16 + nfrag];
      Bf[kt][nt].y = expT[kt * 4 + khalf + 1][nt * 16 + nfrag];
    }

  // alpha0 = start + feats[:,0,:]
  for (int idx = lane; idx < 16 * LSZ; idx += 32) {
    int r = idx / LSZ, j = idx % LSZ;
    alpha[r][j] = start_t[j] + gbase[(size_t)r * TSZ * LSZ + j];
  }
  __syncthreads();

#if USE_TDM
  // Kick off the DMA for the t=1 tile (buffer 1).
  tdm_load_tile((uint32_t)(uintptr_t)&tileT[1][0][0], gbase + 1 * LSZ);
#endif

  for (int t = 1; t < TSZ; ++t) {
    const int cur = t & 1;
#if USE_TDM
    if (t + 1 < TSZ) {
      // Buffer (t+1)&1 was read last iteration; make sure those LDS reads
      // retired before the TDM engine overwrites it, then issue next tile.
      asm volatile("s_wait_dscnt 0x0" ::: "memory");
      tdm_load_tile((uint32_t)(uintptr_t)&tileT[(t + 1) & 1][0][0],
                    gbase + (size_t)(t + 1) * LSZ);
      __builtin_amdgcn_s_wait_tensorcnt(1);  // in-order: tile t is now in LDS
    } else {
      __builtin_amdgcn_s_wait_tensorcnt(0);
    }
    asm volatile("" ::: "memory");           // TDM wrote LDS behind the compiler
#else
    for (int idx = lane; idx < 16 * LSZ; idx += 32) {
      int r = idx / LSZ, j = idx % LSZ;
      tileT[cur][r][j] = gbase[(size_t)r * TSZ * LSZ + (size_t)t * LSZ + j];
    }
    if (t + 1 < TSZ) {
      const int pr = lane & 15, pj = (lane >> 4) * 32;
      __builtin_prefetch(&gbase[(size_t)pr * TSZ * LSZ + (size_t)(t + 1) * LSZ + pj], 0, 1);
    }
#endif
    if (lane < 16) mk[lane] = mask[(size_t)(b0 + lane) * TSZ + t];
    __syncthreads();

    // per-row max for the logsumexp rescale
    if (lane < 16) {
      float m = alpha[lane][0];
#pragma unroll 8
      for (int i = 1; i < LSZ; ++i) m = fmaxf(m, alpha[lane][i]);
      mrow[lane] = m;
    }
    __syncthreads();

    // P = exp(alpha - m)
    for (int idx = lane; idx < 16 * LSZ; idx += 32) {
      int r = idx / LSZ, i = idx % LSZ;
      P[r][i] = __expf(alpha[r][i] - mrow[r]);
    }
    __syncthreads();

    // Z = P @ expT : 3 N-tiles, K=48 in 12 steps of 4
    v8f c0 = {}, c1 = {}, c2 = {};
    const int M = lane & 15;
#pragma unroll
    for (int kt = 0; kt < 12; ++kt) {
      v2f a;
      a.x = P[M][kt * 4 + khalf];
      a.y = P[M][kt * 4 + khalf + 1];
      c0 = __builtin_amdgcn_wmma_f32_16x16x4_f32(false, a, false, Bf[kt][0],
                                                 (short)0, c0, false, false);
      c1 = __builtin_amdgcn_wmma_f32_16x16x4_f32(false, a, false, Bf[kt][1],
                                                 (short)0, c1, false, false);
      c2 = __builtin_amdgcn_wmma_f32_16x16x4_f32(false, a, false, Bf[kt][2],
                                                 (short)0, c2, false, false);
    }
    __syncthreads();

    // alpha' = tile + m + log(Z); C/D layout: VGPR e -> M = e (+8 for hi lanes)
    const volatile float* tile = &tileT[cur][0][0];  // volatile: DMA-written
    const int mbase = (lane >> 4) << 3;
    const int nj = lane & 15;
#pragma unroll
    for (int e = 0; e < 8; ++e) {
      int r = mbase + e;
      if (mk[r]) {
        alpha[r][nj]      = tile[r * LSZ + nj]      + mrow[r] + __logf(c0[e]);
        alpha[r][16 + nj] = tile[r * LSZ + 16 + nj] + mrow[r] + __logf(c1[e]);
        alpha[r][32 + nj] = tile[r * LSZ + 32 + nj] + mrow[r] + __logf(c2[e]);
      }
    }
    __syncthreads();
  }

  // log_Z = logsumexp(alpha + end)
  if (lane < 16) {
    float m = alpha[lane][0] + end_t[0];
#pragma unroll 8
    for (int j = 1; j < LSZ; ++j) m = fmaxf(m, alpha[lane][j] + end_t[j]);
    float s = 0.f;
#pragma unroll 8
    for (int j = 0; j < LSZ; ++j) s += __expf(alpha[lane][j] + end_t[j] - m);
    logZ[b0 + lane] = m + __logf(s);
  }
}

// ---------------------------------------------------------------------------
// Kernel B: Viterbi forward (max-plus; VALU). Thread = (row, label j).
// Writes u8 backpointers bp[b][t-1][j] and per-row argmax of final delta.
// ---------------------------------------------------------------------------
__global__ __launch_bounds__(VROWS * LSZ) void crf_viterbi_fwd(
    const float* __restrict__ feats, const unsigned char* __restrict__ mask,
    const float* __restrict__ trans, const float* __restrict__ start_t,
    const float* __restrict__ end_t,
    unsigned char* __restrict__ bp, int* __restrict__ last)
{
  __shared__ float Tt[LSZ][LPAD];        // Tt[j][i] = trans[i][j]
  __shared__ float dlds[VROWS][LPAD];
  __shared__ float vbuf[VROWS][LPAD];

  const int tid = threadIdx.x;
  const int bl = tid / LSZ;
  const int j  = tid % LSZ;
  const int b  = blockIdx.x * VROWS + bl;

  for (int idx = tid; idx < LSZ * LSZ; idx += VROWS * LSZ) {
    int i = idx / LSZ, jj = idx % LSZ;
    Tt[jj][i] = trans[idx];
  }
  dlds[bl][j] = start_t[j] + feats[(size_t)b * TSZ * LSZ + j];
  __syncthreads();

  const size_t bpbase = (size_t)b * (TSZ - 1) * LSZ;
  for (int t = 1; t < TSZ; ++t) {
    float best = dlds[bl][0] + Tt[j][0];
    int arg = 0;
#pragma unroll 8
    for (int i = 1; i < LSZ; ++i) {
      float cand = dlds[bl][i] + Tt[j][i];
      if (cand > best) { best = cand; arg = i; }   // strict > keeps first (argmax)
    }
    float emit = feats[(size_t)b * TSZ * LSZ + (size_t)t * LSZ + j];
    unsigned char mv = mask[(size_t)b * TSZ + t];
    if (t + 1 < TSZ)
      __builtin_prefetch(&feats[(size_t)b * TSZ * LSZ + (size_t)(t + 1) * LSZ + j], 0, 1);
    __syncthreads();
    unsigned char bpv;
    if (mv) { dlds[bl][j] = best + emit; bpv = (unsigned char)arg; }
    else    { bpv = (unsigned char)j; }            // identity bp when masked
    bp[bpbase + (size_t)(t - 1) * LSZ + j] = bpv;
    __syncthreads();
  }

  vbuf[bl][j] = dlds[bl][j] + end_t[j];
  __syncthreads();
  if (j == 0) {
    float bst = vbuf[bl][0]; int a = 0;
#pragma unroll 8
    for (int i = 1; i < LSZ; ++i)
      if (vbuf[bl][i] > bst) { bst = vbuf[bl][i]; a = i; }
    last[b] = a;
  }
}

// ---------------------------------------------------------------------------
// Kernel C: gold-path score + lengths; one wave per batch row.
// ---------------------------------------------------------------------------
__global__ __launch_bounds__(32) void crf_score(
    const float* __restrict__ feats, const unsigned char* __restrict__ mask,
    const int* __restrict__ tags, const float* __restrict__ trans,
    const float* __restrict__ start_t, const float* __restrict__ end_t,
    float* __restrict__ score, int* __restrict__ lengths)
{
  const int b = blockIdx.x;
  const int lane = threadIdx.x;
  const float* fb = feats + (size_t)b * TSZ * LSZ;
  const int* tb = tags + (size_t)b * TSZ;
  const unsigned char* mb = mask + (size_t)b * TSZ;

  float s = 0.f; int len = 0;
  for (int t = lane; t < TSZ; t += 32) {
    int tg = tb[t];
    len += mb[t] ? 1 : 0;
    if (t == 0) {
      s += start_t[tg] + fb[tg];                   // t=0 terms are unmasked
    } else {
      float mf = mb[t] ? 1.f : 0.f;
      int tp = tb[t - 1];
      s += (fb[(size_t)t * LSZ + tg] + trans[tp * LSZ + tg]) * mf;
    }
  }
  for (int off = 16; off > 0; off >>= 1) {
    s += __shfl_down(s, off, 32);
    len += __shfl_down(len, off, 32);
  }
  if (lane == 0) {
    score[b] = s + end_t[tb[len - 1]];
    lengths[b] = len;
  }
}

// ---------------------------------------------------------------------------
// Kernel D: loss = sum_b (logZ[b] - score[b])  -> d_out[0]
// ---------------------------------------------------------------------------
__global__ __launch_bounds__(256) void crf_loss(
    const float* __restrict__ logZ, const float* __restrict__ score,
    float* __restrict__ out)
{
  __shared__ float red[256];
  float s = 0.f;
  for (int b = threadIdx.x; b < BSZ; b += 256) s += logZ[b] - score[b];
  red[threadIdx.x] = s;
  __syncthreads();
  for (int k = 128; k > 0; k >>= 1) {
    if (threadIdx.x < k) red[threadIdx.x] += red[threadIdx.x + k];
    __syncthreads();
  }
  if (threadIdx.x == 0) out[0] = red[0];
}

// ---------------------------------------------------------------------------
// Kernel E: backtrace. Bulk-stage 32 bp rows (per batch row) into LDS with
// coalesced b32 loads, then run the serial tag-chase against LDS so the
// dependent chain hits ~LDS latency, not global latency.
// Paths written as float into d_out[1..]; -1 past lengths.
// ---------------------------------------------------------------------------
__global__ __launch_bounds__(64) void crf_backtrace(
    const unsigned char* __restrict__ bp, const int* __restrict__ last,
    const int* __restrict__ lengths, float* __restrict__ paths)
{
  __shared__ __align__(16) unsigned char buf[BT_ROWS][BT_CH * LSZ];
  __shared__ int tagc[BT_ROWS];

  const int tid = threadIdx.x;
  const int rl = tid >> 4;
  const int l16 = tid & 15;
  const int b = blockIdx.x * BT_ROWS + rl;
  const int len = lengths[b];

  if (l16 == 0) {
    int tg = last[b];
    tagc[rl] = tg;
    paths[(size_t)b * TSZ + (TSZ - 1)] = (TSZ - 1 < len) ? (float)tg : -1.f;
  }
  const unsigned char* src = bp + (size_t)b * (TSZ - 1) * LSZ;
  __syncthreads();

  for (int hi = TSZ - 2; hi >= 0; hi -= BT_CH) {
    int klo = hi - (BT_CH - 1); if (klo < 0) klo = 0;
    int ndw = ((hi - klo + 1) * LSZ) >> 2;
    const uint32_t* s32 = (const uint32_t*)(src + (size_t)klo * LSZ);
    uint32_t* d32 = (uint32_t*)buf[rl];
    for (int d = l16; d < ndw; d += 16) d32[d] = s32[d];
    __syncthreads();
    if (l16 == 0) {
      int tg = tagc[rl];
      for (int k = hi; k >= klo; --k) {
        tg = buf[rl][(k - klo) * LSZ + tg];
        paths[(size_t)b * TSZ + k] = (k < len) ? (float)tg : -1.f;
      }
      tagc[rl] = tg;
    }
    __syncthreads();
  }
}

// ---------------------------------------------------------------------------
extern "C" void kernel_launch(void* const* d_in, const int* in_sizes, int n_in,
                              void* d_out, int out_size, void* d_ws, size_t ws_size,
                              hipStream_t stream) {
  const float* feats          = (const float*)d_in[0];
  const unsigned char* mask   = (const unsigned char*)d_in[1]; // jnp bool = 1 byte
  const int* tags             = (const int*)d_in[2];
  const float* trans          = (const float*)d_in[3];
  const float* start_t        = (const float*)d_in[4];
  const float* end_t          = (const float*)d_in[5];
  float* out                  = (float*)d_out; // [0]=loss, [1..]=paths (as float)

  char* ws = (char*)d_ws;
  size_t off = 0;
  unsigned char* bp = (unsigned char*)(ws + off);
  off += ((size_t)BSZ * (TSZ - 1) * LSZ + 255) & ~(size_t)255;
  float* logZ   = (float*)(ws + off); off += (BSZ * sizeof(float) + 255) & ~(size_t)255;
  float* score  = (float*)(ws + off); off += (BSZ * sizeof(float) + 255) & ~(size_t)255;
  int* lengths  = (int*)(ws + off);   off += (BSZ * sizeof(int) + 255) & ~(size_t)255;
  int* last     = (int*)(ws + off);   off += (BSZ * sizeof(int) + 255) & ~(size_t)255;
  (void)in_sizes; (void)n_in; (void)out_size; (void)ws_size;

  crf_forward_wmma<<<BSZ / 16, 32, 0, stream>>>(feats, mask, trans, start_t, end_t, logZ);
  crf_viterbi_fwd<<<BSZ / VROWS, VROWS * LSZ, 0, stream>>>(feats, mask, trans, start_t,
                                                           end_t, bp, last);
  crf_score<<<BSZ, 32, 0, stream>>>(feats, mask, tags, trans, start_t, end_t,
                                    score, lengths);
  crf_loss<<<1, 256, 0, stream>>>(logZ, score, out);
  crf_backtrace<<<BSZ / BT_ROWS, 64, 0, stream>>>(bp, last, lengths, out + 1);
}